// BroadcastRouter_45999099740170
// MI455X (gfx1250) — compile-verified
//
#include <hip/hip_runtime.h>
#include <hip/hip_bf16.h>

typedef __attribute__((ext_vector_type(2))) float v2f;
typedef __attribute__((ext_vector_type(8))) float v8f;

#define RQ    4096   // regions
#define BQ    16     // B
#define DQ    256    // D
#define NQ    16     // neighbors
#define TOPK  4
#define BD    4096   // B*D
#define KTOT  512    // 2*D
#define LDS_STRIDE 260  // 256 + 4 pad -> 64-bank conflict-free ds_load_b64

// ---------------------------------------------------------------------------
// Kernel 1: per-region neighbor similarities + stable top-4 -> sel[R][4]
// One block per region, 16 waves (one wave per neighbor).
// ---------------------------------------------------------------------------
__global__ __launch_bounds__(512) void sims_topk_kernel(
    const float* __restrict__ feats,
    const int*   __restrict__ nbr_idx,
    int*         __restrict__ sel)
{
    const int r    = blockIdx.x;
    const int wave = threadIdx.x >> 5;
    const int lane = threadIdx.x & 31;

    __shared__ float s_sims[NQ];

    const int nbr = nbr_idx[r * NQ + wave];
    const float* fr = feats + (size_t)r   * BD;
    const float* fn = feats + (size_t)nbr * BD;

    float s = 0.f;
    for (int i = lane * 4; i < BD; i += 32 * 4) {
        float4 a = *(const float4*)(fr + i);
        float4 b = *(const float4*)(fn + i);
        s += a.x * b.x + a.y * b.y + a.z * b.z + a.w * b.w;
    }
    #pragma unroll
    for (int off = 16; off > 0; off >>= 1)
        s += __shfl_xor(s, off, 32);
    if (lane == 0) s_sims[wave] = s;     // mean scale omitted: order-invariant
    __syncthreads();

    if (threadIdx.x == 0) {
        unsigned used = 0u;
        for (int k = 0; k < TOPK; ++k) {
            int best = 0;
            float bv = -__builtin_inff();
            for (int j = 0; j < NQ; ++j) {
                if (!((used >> j) & 1u) && s_sims[j] > bv) { bv = s_sims[j]; best = j; }
            }
            used |= 1u << best;
            sel[r * TOPK + k] = nbr_idx[r * NQ + best];
        }
    }
}

// ---------------------------------------------------------------------------
// Kernel 2: fused aggregate + linear mix via V_WMMA_F32_16X16X4_F32.
// One block per region (8 waves). A (16x512) staged in LDS, each wave owns
// two 16-wide N tiles, 128 K-steps of WMMA each.
// ---------------------------------------------------------------------------
__global__ __launch_bounds__(256) void route_mix_kernel(
    const float* __restrict__ bcast,
    const int*   __restrict__ sel,
    const float* __restrict__ mix_w,   // [D, 2D] row-major
    const float* __restrict__ mix_b,   // [D]
    float*       __restrict__ out)     // [R, B, D]
{
    __shared__ float Al[BQ][LDS_STRIDE];   // local half of mixed (K = 0..255)
    __shared__ float Ag[BQ][LDS_STRIDE];   // aggregate half   (K = 256..511)

    const int r = blockIdx.x;
    const int t = threadIdx.x;

    // ---- stage A into LDS: Al = bcast[r], Ag = mean of 4 selected rows ----
    const int s0 = sel[r * TOPK + 0];
    const int s1 = sel[r * TOPK + 1];
    const int s2 = sel[r * TOPK + 2];
    const int s3 = sel[r * TOPK + 3];
    const float* bl = bcast + (size_t)r  * BD;
    const float* b0 = bcast + (size_t)s0 * BD;
    const float* b1 = bcast + (size_t)s1 * BD;
    const float* b2 = bcast + (size_t)s2 * BD;
    const float* b3 = bcast + (size_t)s3 * BD;
    #pragma unroll
    for (int k = 0; k < BQ; ++k) {
        const int i = k * DQ + t;                     // coalesced along t
        Al[k][t] = bl[i];
        Ag[k][t] = 0.25f * (b0[i] + b1[i] + b2[i] + b3[i]);
    }
    __syncthreads();

    const int wave = t >> 5;
    const int lane = t & 31;
    const int lo   = lane & 15;       // A: M row / B: N col / D: N col
    const int hi   = lane >> 4;       // K sub-select, D: M+8 select

    #pragma unroll
    for (int tt = 0; tt < 2; ++tt) {
        const int nt = wave + tt * 8;                 // N tile index (0..15)
        const int n  = nt * 16 + lo;                  // output column
        const float* wrow = mix_w + (size_t)n * KTOT + 2 * hi;

        v8f acc = {};

        // local half: k_global = 0..255
        #pragma unroll 8
        for (int ks = 0; ks < 64; ++ks) {
            v2f a = *(const v2f*)(&Al[lo][ks * 4 + 2 * hi]);  // ds_load_b64
            v2f b = *(const v2f*)(wrow + ks * 4);             // global b64
            acc = __builtin_amdgcn_wmma_f32_16x16x4_f32(
                false, a, false, b, (short)0, acc, false, false);
        }
        // aggregate half: k_global = 256..511
        #pragma unroll 8
        for (int ks = 0; ks < 64; ++ks) {
            v2f a = *(const v2f*)(&Ag[lo][ks * 4 + 2 * hi]);
            v2f b = *(const v2f*)(wrow + DQ + ks * 4);
            acc = __builtin_amdgcn_wmma_f32_16x16x4_f32(
                false, a, false, b, (short)0, acc, false, false);
        }

        const float bias = mix_b[n];
        float* op = out + (size_t)r * BD + n;
        #pragma unroll
        for (int v = 0; v < 8; ++v) {
            const int m = v + 8 * hi;                 // C/D layout: M = v (+8 for hi lanes)
            op[m * DQ] = acc[v] + bias;
        }
    }
}

// ---------------------------------------------------------------------------
extern "C" void kernel_launch(void* const* d_in, const int* in_sizes, int n_in,
                              void* d_out, int out_size, void* d_ws, size_t ws_size,
                              hipStream_t stream) {
    const float* bcast = (const float*)d_in[0];
    const float* feats = (const float*)d_in[1];
    const int*   nbr   = (const int*)  d_in[2];
    const float* mix_w = (const float*)d_in[3];
    const float* mix_b = (const float*)d_in[4];
    float* out = (float*)d_out;
    int*   sel = (int*)d_ws;                       // R*4 ints of scratch

    hipLaunchKernelGGL(sims_topk_kernel, dim3(RQ), dim3(512), 0, stream,
                       feats, nbr, sel);
    hipLaunchKernelGGL(route_mix_kernel, dim3(RQ), dim3(256), 0, stream,
                       bcast, sel, mix_w, mix_b, out);
}